// HDPHMM_70463233458519
// MI455X (gfx1250) — compile-verified
//
#include <hip/hip_runtime.h>
#include <math.h>

typedef float v2f __attribute__((ext_vector_type(2)));
typedef float v8f __attribute__((ext_vector_type(8)));

#define T_LEN   131072
#define KST     64
#define CHUNK   256
#define NCHUNK  (T_LEN / CHUNK)     // 512
#define HMM_EPS 1e-10f
#define LOG2PI  1.8378770664093453f

// workspace layout (float offsets)
#define OFF_P     0                       // 64x64 transition matrix
#define OFF_W     4096                    // 16x64 emission weight matrix
#define OFF_CVEC  5120                    // 64
#define OFF_BW    5184                    // 64 stick-breaking weights
#define OFF_VB    5248                    // NCHUNK*64 fwd boundary vectors
#define OFF_BB    (OFF_VB + NCHUNK*KST)   // NCHUNK*64 bwd boundary vectors
#define OFF_EM    (OFF_BB + NCHUNK*KST)   // T*K emissions
#define OFF_G     (OFF_EM + T_LEN*KST)    // NCHUNK*4096 chunk matrices
// total ~10.56M floats (~42 MB)

__device__ inline v8f wmma_f32_4(v2f a, v2f b, v8f c) {
  return __builtin_amdgcn_wmma_f32_16x16x4_f32(false, a, false, b, (short)0, c,
                                               false, false);
}

__device__ inline float wave_sum(float v) {
#pragma unroll
  for (int m = 16; m >= 1; m >>= 1) v += __shfl_xor(v, m, 32);
  return v;
}
__device__ inline float wave_max(float v) {
#pragma unroll
  for (int m = 16; m >= 1; m >>= 1) v = fmaxf(v, __shfl_xor(v, m, 32));
  return v;
}
// block of 64 threads (2 waves)
__device__ inline float block64_sum(float v, float* red2) {
  int wv = threadIdx.x >> 5, ln = threadIdx.x & 31;
  v = wave_sum(v);
  if (ln == 0) red2[wv] = v;
  __syncthreads();
  return red2[0] + red2[1];
}

// ---------- K0: transition softmax, emission weights, stick-breaking ----------
__global__ void k_prep(const float* __restrict__ beta_logits,
                       const float* __restrict__ pi_logits,
                       const float* __restrict__ means,
                       const float* __restrict__ log_vars,
                       float* __restrict__ ws) {
  int k = threadIdx.x;           // 64 threads
  float* P    = ws + OFF_P;
  float* W    = ws + OFF_W;
  float* cvec = ws + OFF_CVEC;
  float* bw   = ws + OFF_BW;
  const float* row = pi_logits + k * 64;
  float m = row[0];
  for (int j = 1; j < 64; ++j) m = fmaxf(m, row[j]);
  float s = 0.0f;
  for (int j = 0; j < 64; ++j) s += expf(row[j] - m);
  float invs = 1.0f / s;
  for (int j = 0; j < 64; ++j) P[k * 64 + j] = expf(row[j] - m) * invs;
  float csum = 0.0f;
  for (int f = 0; f < 8; ++f) {
    float lv = log_vars[k * 8 + f];
    float iv = expf(-lv);
    float mu = means[k * 8 + f];
    W[f * 64 + k]       = -0.5f * iv;   // coeff of x^2
    W[(8 + f) * 64 + k] = mu * iv;      // coeff of x
    csum += mu * mu * iv + lv;
  }
  cvec[k] = -0.5f * (csum + 8.0f * LOG2PI);
  if (k == 0) {
    float cp = 1.0f;
    for (int i = 0; i < 64; ++i) {
      float b = 1.0f / (1.0f + expf(-beta_logits[i]));
      bw[i] = b * cp;
      cp *= (1.0f - b);
    }
  }
}

// ---------- K1: emissions via WMMA: em = exp([x^2, x] @ W + c) ----------
__global__ void __launch_bounds__(256) k_em(const float* __restrict__ obs,
                                            float* __restrict__ ws) {
  __shared__ v2f Wp[8 * 64];   // W K-pairs in B layout
  __shared__ float cv[KST];
  const float* W    = ws + OFF_W;
  const float* cvec = ws + OFF_CVEC;
  float* em         = ws + OFF_EM;
  int tid = threadIdx.x;
  for (int idx = tid; idx < 8 * 64; idx += 256) {
    int kp = idx >> 6, n = idx & 63;
    v2f w2; w2[0] = W[(2 * kp) * 64 + n]; w2[1] = W[(2 * kp + 1) * 64 + n];
    Wp[idx] = w2;
  }
  if (tid < 64) cv[tid] = cvec[tid];
  __syncthreads();
  int wave = tid >> 5, lane = tid & 31;
  int hi = lane >> 4, lm = lane & 15, koff = hi * 2;
  int t0 = blockIdx.x * 128 + wave * 16;   // 8 waves x 16 rows
  const float* orow = obs + (t0 + lm) * 8;
  // Each lane only needs obs columns {koff, koff+1, koff+4, koff+5}: load the
  // two aligned float2's directly -- no dynamically indexed register array.
  v2f x01 = *(const v2f*)(orow + koff);
  v2f x45 = *(const v2f*)(orow + koff + 4);
  v2f a[4];
  a[0][0] = x01[0] * x01[0]; a[0][1] = x01[1] * x01[1];  // K 0..7  : x^2 part
  a[1][0] = x45[0] * x45[0]; a[1][1] = x45[1] * x45[1];
  a[2] = x01;                                            // K 8..15 : x part
  a[3] = x45;
#pragma unroll
  for (int nb = 0; nb < 4; ++nb) {
    v8f acc = {0, 0, 0, 0, 0, 0, 0, 0};
#pragma unroll
    for (int kb = 0; kb < 4; ++kb) {
      v2f b = Wp[(kb * 2 + hi) * 64 + nb * 16 + lm];
      acc = wmma_f32_4(a[kb], b, acc);
    }
    int col = nb * 16 + lm;
    float cadd = cv[col];
#pragma unroll
    for (int r = 0; r < 8; ++r) {
      int t = t0 + r + 8 * hi;
      em[t * 64 + col] = expf(acc[r] + cadd);
    }
  }
}

// ---------- K2: chunk transfer matrices G_c = prod_t P*diag(e_t) (WMMA) -----
__global__ void __launch_bounds__(128) k_passA(float* __restrict__ ws) {
  __shared__ float Rs[4][16 * 64];   // per-wave private 16-row slab
  __shared__ v2f Pp[32 * 64];        // P K-pairs in B layout
  __shared__ float lsh[4];
  const float* P  = ws + OFF_P;
  const float* em = ws + OFF_EM;
  float* G        = ws + OFF_G;
  int tid = threadIdx.x;
  int c = blockIdx.x;
  for (int idx = tid; idx < 32 * 64; idx += 128) {
    int kp = idx >> 6, n = idx & 63;
    v2f p2; p2[0] = P[(2 * kp) * 64 + n]; p2[1] = P[(2 * kp + 1) * 64 + n];
    Pp[idx] = p2;
  }
  for (int idx = tid; idx < 4096; idx += 128) {   // identity init
    int w = idx >> 10, rem = idx & 1023;
    int m = rem >> 6, k = rem & 63;
    Rs[w][rem] = (16 * w + m == k) ? 1.0f : 0.0f;
  }
  __syncthreads();
  int wave = tid >> 5, lane = tid & 31;
  int hi = lane >> 4, lm = lane & 15, koff = 2 * hi;
  float inv = 1.0f, def = 0.0f;          // running log2 scale deficit of Rs
  int tb = (c == 0) ? 1 : c * CHUNK;     // chunk 0 excludes t=0 (handled by a0)
  int te = c * CHUNK + CHUNK;
  v8f acc[4];
  for (int t = tb; t < te; ++t) {
    def -= log2f(inv);
#pragma unroll
    for (int nb = 0; nb < 4; ++nb) {
      v8f s = {0, 0, 0, 0, 0, 0, 0, 0};
#pragma unroll
      for (int kb = 0; kb < 16; ++kb) {
        v2f av = *(const v2f*)&Rs[wave][lm * 64 + kb * 4 + koff];
        v2f bv = Pp[(kb * 2 + hi) * 64 + nb * 16 + lm];
        s = wmma_f32_4(av, bv, s);
      }
      acc[nb] = s;
    }
    float lmax = 0.0f;
#pragma unroll
    for (int nb = 0; nb < 4; ++nb) {
      float e = em[t * 64 + nb * 16 + lm];
      float sc = e * inv;
#pragma unroll
      for (int r = 0; r < 8; ++r) {
        float v = acc[nb][r] * sc;
        acc[nb][r] = v;
        lmax = fmaxf(lmax, v);
      }
    }
    lmax = wave_max(lmax);
    lmax = fmaxf(lmax, 1e-37f);
    inv = 1.0f / lmax;                   // applied next step; log2 folded into def
#pragma unroll
    for (int nb = 0; nb < 4; ++nb)
#pragma unroll
      for (int r = 0; r < 8; ++r)
        Rs[wave][(r + 8 * hi) * 64 + nb * 16 + lm] = acc[nb][r];
  }
  if (lane == 0) lsh[wave] = def;
  __syncthreads();
  float dmax = fmaxf(fmaxf(lsh[0], lsh[1]), fmaxf(lsh[2], lsh[3]));
  float fac = exp2f(def - dmax);         // reconcile per-slab row scales
  float* Gc = G + c * 4096;
#pragma unroll
  for (int nb = 0; nb < 4; ++nb)
#pragma unroll
    for (int r = 0; r < 8; ++r)
      Gc[(16 * wave + r + 8 * hi) * 64 + nb * 16 + lm] = acc[nb][r] * fac;
}

// ---------- K3: boundary scans (fwd: v <- v G_c ; bwd: w <- G_c w) ----------
__global__ void k_bound(float* __restrict__ ws) {
  __shared__ float shv[64];
  __shared__ float red2[2];
  const float* G  = ws + OFF_G;
  const float* em = ws + OFF_EM;
  const float* bw = ws + OFF_BW;
  float* vb = ws + OFF_VB;
  float* bb = ws + OFF_BB;
  int j = threadIdx.x;   // 64 threads
  if (blockIdx.x == 0) {
    float a = bw[j] * em[j];
    float s = block64_sum(a, red2);
    a /= (s + HMM_EPS);
    __syncthreads();
    shv[j] = a;
    __syncthreads();
    for (int c = 0; c < NCHUNK; ++c) {
      const float* Gc = G + c * 4096;
      float d = 0.0f;
      for (int i = 0; i < 64; ++i) d += shv[i] * Gc[i * 64 + j];
      float s2 = block64_sum(d, red2);
      float an = d / (s2 + HMM_EPS);
      __syncthreads();
      shv[j] = an;
      vb[c * 64 + j] = an;           // alpha at end of chunk c
      __syncthreads();
    }
  } else {
    bb[(NCHUNK - 1) * 64 + j] = 1.0f;  // beta[T-1] = ones (unnormalized)
    shv[j] = 1.0f;
    __syncthreads();
    for (int c = NCHUNK - 1; c >= 1; --c) {
      const float* Gc = G + c * 4096;
      float d = 0.0f;
      for (int jj = 0; jj < 64; ++jj) d += Gc[j * 64 + jj] * shv[jj];
      float s2 = block64_sum(d, red2);
      float bn = d / (s2 + HMM_EPS);
      __syncthreads();
      shv[j] = bn;
      bb[(c - 1) * 64 + j] = bn;     // beta at end of chunk c-1
      __syncthreads();
    }
  }
}

// ---------- K4: per-chunk replay, writes alpha/beta with exact ref normalization
__global__ void __launch_bounds__(64) k_passB(float* __restrict__ ws,
                                              float* __restrict__ out) {
  __shared__ float sh[64];
  __shared__ float red2[2];
  const float* P  = ws + OFF_P;
  const float* em = ws + OFF_EM;
  const float* bw = ws + OFF_BW;
  const float* vb = ws + OFF_VB;
  const float* bb = ws + OFF_BB;
  float* alpha = out;
  float* beta  = out + (size_t)T_LEN * KST;
  int j = threadIdx.x;
  int dir = blockIdx.x / NCHUNK;
  int c = blockIdx.x % NCHUNK;
  if (dir == 0) {
    float pc[64];                      // column j of P
#pragma unroll
    for (int i = 0; i < 64; ++i) pc[i] = P[i * 64 + j];
    float a; int tb;
    if (c == 0) {
      a = bw[j] * em[j];
      float s = block64_sum(a, red2);
      a /= (s + HMM_EPS);
      alpha[j] = a;
      tb = 1;
    } else {
      a = vb[(c - 1) * 64 + j];
      tb = c * CHUNK;
    }
    __syncthreads();
    sh[j] = a;
    __syncthreads();
    int te = c * CHUNK + CHUNK;
    for (int t = tb; t < te; ++t) {
      float d = 0.0f;
#pragma unroll
      for (int i = 0; i < 64; ++i) d += sh[i] * pc[i];
      d *= em[t * 64 + j];
      float s = block64_sum(d, red2);
      float an = d / (s + HMM_EPS);
      alpha[(size_t)t * 64 + j] = an;
      __syncthreads();
      sh[j] = an;
      __syncthreads();
    }
  } else {
    float pr[64];                      // row j of P
#pragma unroll
    for (int jj = 0; jj < 64; ++jj) pr[jj] = P[j * 64 + jj];
    float b = bb[c * 64 + j];
    int thigh = (c + 1) * CHUNK - 1;
    beta[(size_t)thigh * 64 + j] = b;
    sh[j] = b * em[(size_t)thigh * 64 + j];   // tmp = b[t+1]*e[t+1]
    __syncthreads();
    for (int t = thigh - 1; t >= c * CHUNK; --t) {
      float d = 0.0f;
#pragma unroll
      for (int jj = 0; jj < 64; ++jj) d += pr[jj] * sh[jj];
      float s = block64_sum(d, red2);
      float bn = d / (s + HMM_EPS);
      beta[(size_t)t * 64 + j] = bn;
      __syncthreads();
      sh[j] = bn * em[(size_t)t * 64 + j];
      __syncthreads();
    }
  }
}

// ---------- K5: log-likelihood ----------
__global__ void k_ll(const float* __restrict__ out_ro, float* __restrict__ out) {
  __shared__ float red2[2];
  int j = threadIdx.x;
  float v = out_ro[(size_t)(T_LEN - 1) * KST + j];
  float s = block64_sum(v, red2);
  if (j == 0) out[(size_t)2 * T_LEN * KST] = logf(s + HMM_EPS);
}

extern "C" void kernel_launch(void* const* d_in, const int* in_sizes, int n_in,
                              void* d_out, int out_size, void* d_ws, size_t ws_size,
                              hipStream_t stream) {
  (void)in_sizes; (void)n_in; (void)out_size; (void)ws_size;
  const float* obs         = (const float*)d_in[0];
  const float* beta_logits = (const float*)d_in[1];
  const float* pi_logits   = (const float*)d_in[2];
  const float* means       = (const float*)d_in[3];
  const float* log_vars    = (const float*)d_in[4];
  float* ws  = (float*)d_ws;
  float* out = (float*)d_out;

  k_prep<<<1, 64, 0, stream>>>(beta_logits, pi_logits, means, log_vars, ws);
  k_em<<<T_LEN / 128, 256, 0, stream>>>(obs, ws);
  k_passA<<<NCHUNK, 128, 0, stream>>>(ws);
  k_bound<<<2, 64, 0, stream>>>(ws);
  k_passB<<<2 * NCHUNK, 64, 0, stream>>>(ws, out);
  k_ll<<<1, 64, 0, stream>>>(out, out);
}